// grud_model_35622458753091
// MI455X (gfx1250) — compile-verified
//
#include <hip/hip_runtime.h>
#include <hip/hip_bf16.h>
#include <math.h>

// Problem constants (reference: B=64, T=512, F=128, H=256, O=1)
#define B_ 64
#define T_ 512
#define F_ 128
#define H_ 256

typedef __attribute__((ext_vector_type(2))) float v2f;
typedef __attribute__((ext_vector_type(8))) float v8f;

__device__ __forceinline__ v8f wmma4(v2f a, v2f b, v8f c) {
  // D = A(16x4,f32) * B(4x16,f32) + C(16x16,f32)
  return __builtin_amdgcn_wmma_f32_16x16x4_f32(false, a, false, b, (short)0, c,
                                               false, false);
}

__device__ __forceinline__ float sigmoidf_(float x) {
  return 1.0f / (1.0f + expf(-x));
}

__device__ __forceinline__ v8f vzero8() {
  v8f z = {0.f, 0.f, 0.f, 0.f, 0.f, 0.f, 0.f, 0.f};
  return z;
}

// ---------------------------------------------------------------------------
// K0: repack a (K x N) row-major weight matrix into WMMA-B-fragment order so
// each lane loads its two f32 B elements (rows k+2*lh, k+2*lh+1 at column c)
// with ONE aligned global_load_b64:
//   out[(((k>>2)*2 + ((k&3)>>1)) * N + c) * 2 + (k&1)] = in[k*N + c]
// GEMM-side fragment address: out + ((k>>1) + lh)*2N + 2*col
// ---------------------------------------------------------------------------
__global__ __launch_bounds__(256) void k_repack(const float* __restrict__ in,
                                                float* __restrict__ out,
                                                int K, int N) {
  const int idx = blockIdx.x * 256 + threadIdx.x;
  if (idx >= K * N) return;
  const int k = idx / N, c = idx - k * N;
  const size_t o = ((size_t)((k >> 2) * 2 + ((k & 3) >> 1)) * N + c) * 2 + (k & 1);
  out[o] = in[idx];
}

// ---------------------------------------------------------------------------
// K1: gamma = exp(-relu(delta @ Wg + bg)), rows time-major r = t*B + b.
// A (delta) is stored (B,T,F) so row r maps to delta[(r&63)*T + (r>>6)].
// 2x-unrolled ping-pong k-loop (no register rotation copies), packed B.
// grid: (N/64, 32768/128), block 256 (8 waves, 1 m-tile x 4 n-tiles each)
// ---------------------------------------------------------------------------
__global__ __launch_bounds__(256) void k_gamma(const float* __restrict__ delta,
                                               const float* __restrict__ Wgp,
                                               const float* __restrict__ bg,
                                               float* __restrict__ out,
                                               int N) {
  const int lane = threadIdx.x & 31, w = threadIdx.x >> 5;
  const int lh = lane >> 4, lm = lane & 15;
  const int row0 = blockIdx.y * 128 + w * 16;
  const int col0 = blockIdx.x * 64;

  v8f acc[4];
#pragma unroll
  for (int j = 0; j < 4; ++j) acc[j] = vzero8();

  const int r = row0 + lm;
  const float* arow = delta + ((size_t)(r & 63) * T_ + (r >> 6)) * F_ + 2 * lh;
  const float* bbase = Wgp + (size_t)lh * 2 * N + 2 * (col0 + lm);

  v2f a0, a1, b0[4], b1[4];
  a0 = *(const v2f*)(arow);
#pragma unroll
  for (int j = 0; j < 4; ++j) b0[j] = *(const v2f*)(bbase + j * 32);

  int k = 0;
  for (; k + 8 < F_; k += 8) {
    const float* bn1 = bbase + (size_t)((k + 4) >> 1) * (2 * N);
    a1 = *(const v2f*)(arow + k + 4);
#pragma unroll
    for (int j = 0; j < 4; ++j) b1[j] = *(const v2f*)(bn1 + j * 32);
#pragma unroll
    for (int j = 0; j < 4; ++j) acc[j] = wmma4(a0, b0[j], acc[j]);
    const float* bn0 = bbase + (size_t)((k + 8) >> 1) * (2 * N);
    a0 = *(const v2f*)(arow + k + 8);
#pragma unroll
    for (int j = 0; j < 4; ++j) b0[j] = *(const v2f*)(bn0 + j * 32);
#pragma unroll
    for (int j = 0; j < 4; ++j) acc[j] = wmma4(a1, b1[j], acc[j]);
  }
  {  // k == F_-8: buffers hold block k; fetch k+4 then drain both
    const float* bn1 = bbase + (size_t)((k + 4) >> 1) * (2 * N);
    a1 = *(const v2f*)(arow + k + 4);
#pragma unroll
    for (int j = 0; j < 4; ++j) b1[j] = *(const v2f*)(bn1 + j * 32);
#pragma unroll
    for (int j = 0; j < 4; ++j) acc[j] = wmma4(a0, b0[j], acc[j]);
#pragma unroll
    for (int j = 0; j < 4; ++j) acc[j] = wmma4(a1, b1[j], acc[j]);
  }

#pragma unroll
  for (int j = 0; j < 4; ++j) {
    const int col = col0 + j * 16 + lm;
    const float bv = bg[col];
#pragma unroll
    for (int i = 0; i < 8; ++i) {
      const int row = row0 + i + 8 * lh;
      float v = acc[j][i] + bv;
      v = expf(-fmaxf(v, 0.0f));
      out[(size_t)row * N + col] = v;
    }
  }
}

// ---------------------------------------------------------------------------
// K2: elementwise x_last scan + x_hat.  One thread per (b,f), loops over t.
// x_hat written time-major (t*B + b, f).
// ---------------------------------------------------------------------------
__global__ __launch_bounds__(256) void k_xhat(const float* __restrict__ x,
                                              const float* __restrict__ mask,
                                              const float* __restrict__ gx,
                                              const float* __restrict__ xmean,
                                              float* __restrict__ xhat) {
  const int idx = blockIdx.x * 256 + threadIdx.x;  // 0..B*F-1
  const int b = idx >> 7, f = idx & 127;
  const float xm = xmean[f];
  float xl = 0.0f;
  for (int t = 0; t < T_; ++t) {
    const size_t gio = ((size_t)b * T_ + t) * F_ + f;      // (B,T,F)
    const size_t tmo = ((size_t)t * B_ + b) * F_ + f;      // time-major
    const float m = mask[gio];
    const float xv = x[gio];
    const float g = gx[tmo];
    xhat[tmo] = m * xv + (1.0f - m) * (g * xl + (1.0f - g) * xm);
    xl = m * xv + (1.0f - m) * xl;
  }
}

// ---------------------------------------------------------------------------
// K3: pre = x_hat @ W_x + mask @ W_m + b_g   (32768 x 768, K=128 twice)
// ---------------------------------------------------------------------------
__global__ __launch_bounds__(256) void k_pre(const float* __restrict__ xhat,
                                             const float* __restrict__ mask,
                                             const float* __restrict__ Wxp,
                                             const float* __restrict__ Wmp,
                                             const float* __restrict__ bgv,
                                             float* __restrict__ pre) {
  const int lane = threadIdx.x & 31, w = threadIdx.x >> 5;
  const int lh = lane >> 4, lm = lane & 15;
  const int row0 = blockIdx.y * 128 + w * 16;
  const int col0 = blockIdx.x * 64;
  const int N = 3 * H_;

  v8f acc[4];
#pragma unroll
  for (int j = 0; j < 4; ++j) acc[j] = vzero8();

  const int r = row0 + lm;
  const float* a0p = xhat + (size_t)r * F_ + 2 * lh;  // time-major contiguous
  const float* a1p = mask + ((size_t)(r & 63) * T_ + (r >> 6)) * F_ + 2 * lh;
  const size_t bfrag = (size_t)lh * 2 * N + 2 * (col0 + lm);

#pragma unroll
  for (int ph = 0; ph < 2; ++ph) {
    const float* ap = (ph == 0) ? a0p : a1p;
    const float* bbase = ((ph == 0) ? Wxp : Wmp) + bfrag;
    v2f a0, a1, b0[4], b1[4];
    a0 = *(const v2f*)(ap);
#pragma unroll
    for (int j = 0; j < 4; ++j) b0[j] = *(const v2f*)(bbase + j * 32);
    int k = 0;
    for (; k + 8 < F_; k += 8) {
      const float* bn1 = bbase + (size_t)((k + 4) >> 1) * (2 * N);
      a1 = *(const v2f*)(ap + k + 4);
#pragma unroll
      for (int j = 0; j < 4; ++j) b1[j] = *(const v2f*)(bn1 + j * 32);
#pragma unroll
      for (int j = 0; j < 4; ++j) acc[j] = wmma4(a0, b0[j], acc[j]);
      const float* bn0 = bbase + (size_t)((k + 8) >> 1) * (2 * N);
      a0 = *(const v2f*)(ap + k + 8);
#pragma unroll
      for (int j = 0; j < 4; ++j) b0[j] = *(const v2f*)(bn0 + j * 32);
#pragma unroll
      for (int j = 0; j < 4; ++j) acc[j] = wmma4(a1, b1[j], acc[j]);
    }
    {
      const float* bn1 = bbase + (size_t)((k + 4) >> 1) * (2 * N);
      a1 = *(const v2f*)(ap + k + 4);
#pragma unroll
      for (int j = 0; j < 4; ++j) b1[j] = *(const v2f*)(bn1 + j * 32);
#pragma unroll
      for (int j = 0; j < 4; ++j) acc[j] = wmma4(a0, b0[j], acc[j]);
#pragma unroll
      for (int j = 0; j < 4; ++j) acc[j] = wmma4(a1, b1[j], acc[j]);
    }
  }
#pragma unroll
  for (int j = 0; j < 4; ++j) {
    const int col = col0 + j * 16 + lm;
    const float bv = bgv[col];
#pragma unroll
    for (int i = 0; i < 8; ++i) {
      const int row = row0 + i + 8 * lh;
      pre[(size_t)row * N + col] = acc[j][i] + bv;
    }
  }
}

// ---------------------------------------------------------------------------
// K4: persistent GRU-D recurrence.  1 workgroup, 1024 threads (32 waves).
// LDS: h (64x256 f32) + r (64x256 f32) = 128KB.
// Per step: h *= gamma_h; zr = sigmoid(pre_zr + h@U_zr) (z kept in regs,
// r -> LDS); n = tanh(pre_n + (r*h)@U_n); h = (1-z)*h + z*n; store hs[t].
// ---------------------------------------------------------------------------
__global__ __launch_bounds__(1024) void k_grud_seq(const float* __restrict__ pre,
                                                   const float* __restrict__ gh_all,
                                                   const float* __restrict__ Uzrp,
                                                   const float* __restrict__ Unp,
                                                   float* __restrict__ hs) {
  extern __shared__ float smem[];
  float* sh = smem;             // 64 x 256 (h, decayed within step)
  float* sr = smem + B_ * H_;   // 64 x 256 (r gate)

  const int tid = threadIdx.x;
  const int lane = tid & 31, w = tid >> 5;
  const int lh = lane >> 4, lm = lane & 15;
  const int m_t = w >> 3, oct = w & 7;
  const int row0 = m_t * 16;

  for (int e = tid; e < B_ * H_; e += 1024) sh[e] = 0.0f;
  __syncthreads();

  // per-wave packed-B fragment bases (columns fixed across t)
  const int c0 = (oct * 2) * 16 + lm;        // z col tile 0 (and n col tile 0)
  const float* uz_base = Uzrp + (size_t)lh * 2 * 512 + 2 * c0;   // N=512
  const float* un_base = Unp + (size_t)lh * 2 * 256 + 2 * c0;    // N=256

  for (int t = 0; t < T_; ++t) {
    // --- decay: h = gamma_h * h ---
    const float* ght = gh_all + (size_t)t * (B_ * H_);
    for (int e = tid; e < B_ * H_; e += 1024) sh[e] *= ght[e];
    __syncthreads();

    const float* pre_t = pre + (size_t)t * (B_ * 3 * H_);

    // --- GEMM1: zr = h @ U_zr (64x512, K=256) ---
    v8f zacc[2], racc[2];
    zacc[0] = zacc[1] = racc[0] = racc[1] = vzero8();
    const float* arow = sh + (row0 + lm) * H_ + 2 * lh;

    v2f a0, a1, bz0[2], br0[2], bz1[2], br1[2];
    a0 = *(const v2f*)(arow);
#pragma unroll
    for (int j = 0; j < 2; ++j) {
      bz0[j] = *(const v2f*)(uz_base + j * 32);
      br0[j] = *(const v2f*)(uz_base + j * 32 + 512);  // +256 cols * 2
    }
    int k = 0;
    for (; k + 8 < H_; k += 8) {
      const float* bn1 = uz_base + (size_t)((k + 4) >> 1) * 1024;  // 2*N
      a1 = *(const v2f*)(arow + k + 4);
#pragma unroll
      for (int j = 0; j < 2; ++j) {
        bz1[j] = *(const v2f*)(bn1 + j * 32);
        br1[j] = *(const v2f*)(bn1 + j * 32 + 512);
      }
#pragma unroll
      for (int j = 0; j < 2; ++j) {
        zacc[j] = wmma4(a0, bz0[j], zacc[j]);
        racc[j] = wmma4(a0, br0[j], racc[j]);
      }
      const float* bn0 = uz_base + (size_t)((k + 8) >> 1) * 1024;
      a0 = *(const v2f*)(arow + k + 8);
#pragma unroll
      for (int j = 0; j < 2; ++j) {
        bz0[j] = *(const v2f*)(bn0 + j * 32);
        br0[j] = *(const v2f*)(bn0 + j * 32 + 512);
      }
#pragma unroll
      for (int j = 0; j < 2; ++j) {
        zacc[j] = wmma4(a1, bz1[j], zacc[j]);
        racc[j] = wmma4(a1, br1[j], racc[j]);
      }
    }
    {
      const float* bn1 = uz_base + (size_t)((k + 4) >> 1) * 1024;
      a1 = *(const v2f*)(arow + k + 4);
#pragma unroll
      for (int j = 0; j < 2; ++j) {
        bz1[j] = *(const v2f*)(bn1 + j * 32);
        br1[j] = *(const v2f*)(bn1 + j * 32 + 512);
      }
#pragma unroll
      for (int j = 0; j < 2; ++j) {
        zacc[j] = wmma4(a0, bz0[j], zacc[j]);
        racc[j] = wmma4(a0, br0[j], racc[j]);
      }
#pragma unroll
      for (int j = 0; j < 2; ++j) {
        zacc[j] = wmma4(a1, bz1[j], zacc[j]);
        racc[j] = wmma4(a1, br1[j], racc[j]);
      }
    }

#pragma unroll
    for (int j = 0; j < 2; ++j) {
      const int c = (oct * 2 + j) * 16 + lm;
#pragma unroll
      for (int i = 0; i < 8; ++i) {
        const int m = row0 + i + 8 * lh;
        const float zv = sigmoidf_(zacc[j][i] + pre_t[(size_t)m * 768 + c]);
        zacc[j][i] = zv;  // keep z in registers
        const float rv = sigmoidf_(racc[j][i] + pre_t[(size_t)m * 768 + H_ + c]);
        sr[m * H_ + c] = rv;
      }
    }
    __syncthreads();

    // --- GEMM2: n = tanh(pre_n + (r*h) @ U_n) (64x256, K=256) ---
    v8f nacc[2];
    nacc[0] = nacc[1] = vzero8();
    const int abase = (row0 + lm) * H_ + 2 * lh;
    v2f bu0[2], bu1[2];
#pragma unroll
    for (int j = 0; j < 2; ++j) bu0[j] = *(const v2f*)(un_base + j * 32);
    k = 0;
    for (; k + 8 < H_; k += 8) {
      const float* bn1 = un_base + (size_t)((k + 4) >> 1) * 512;  // 2*N
#pragma unroll
      for (int j = 0; j < 2; ++j) bu1[j] = *(const v2f*)(bn1 + j * 32);
      {
        const int ab = abase + k;
        v2f a; a.x = sr[ab] * sh[ab]; a.y = sr[ab + 1] * sh[ab + 1];
#pragma unroll
        for (int j = 0; j < 2; ++j) nacc[j] = wmma4(a, bu0[j], nacc[j]);
      }
      const float* bn0 = un_base + (size_t)((k + 8) >> 1) * 512;
#pragma unroll
      for (int j = 0; j < 2; ++j) bu0[j] = *(const v2f*)(bn0 + j * 32);
      {
        const int ab = abase + k + 4;
        v2f a; a.x = sr[ab] * sh[ab]; a.y = sr[ab + 1] * sh[ab + 1];
#pragma unroll
        for (int j = 0; j < 2; ++j) nacc[j] = wmma4(a, bu1[j], nacc[j]);
      }
    }
    {
      const float* bn1 = un_base + (size_t)((k + 4) >> 1) * 512;
#pragma unroll
      for (int j = 0; j < 2; ++j) bu1[j] = *(const v2f*)(bn1 + j * 32);
      {
        const int ab = abase + k;
        v2f a; a.x = sr[ab] * sh[ab]; a.y = sr[ab + 1] * sh[ab + 1];
#pragma unroll
        for (int j = 0; j < 2; ++j) nacc[j] = wmma4(a, bu0[j], nacc[j]);
      }
      {
        const int ab = abase + k + 4;
        v2f a; a.x = sr[ab] * sh[ab]; a.y = sr[ab + 1] * sh[ab + 1];
#pragma unroll
        for (int j = 0; j < 2; ++j) nacc[j] = wmma4(a, bu1[j], nacc[j]);
      }
    }
    float hnew[2][8];
#pragma unroll
    for (int j = 0; j < 2; ++j) {
      const int c = (oct * 2 + j) * 16 + lm;
#pragma unroll
      for (int i = 0; i < 8; ++i) {
        const int m = row0 + i + 8 * lh;
        const float nv = tanhf(nacc[j][i] + pre_t[(size_t)m * 768 + 2 * H_ + c]);
        const float zv = zacc[j][i];
        const float hv = sh[m * H_ + c];
        hnew[j][i] = (1.0f - zv) * hv + zv * nv;
      }
    }
    __syncthreads();
    float* hs_t = hs + (size_t)t * (B_ * H_);
#pragma unroll
    for (int j = 0; j < 2; ++j) {
      const int c = (oct * 2 + j) * 16 + lm;
#pragma unroll
      for (int i = 0; i < 8; ++i) {
        const int m = row0 + i + 8 * lh;
        sh[m * H_ + c] = hnew[j][i];
        hs_t[(size_t)m * H_ + c] = hnew[j][i];
      }
    }
    __syncthreads();
  }
}

// ---------------------------------------------------------------------------
// K5: gi = hs @ W_ih + b_ih   (32768 x 768, K=256), A contiguous time-major.
// ---------------------------------------------------------------------------
__global__ __launch_bounds__(256) void k_gi(const float* __restrict__ hsin,
                                            const float* __restrict__ Wihp,
                                            const float* __restrict__ bih,
                                            float* __restrict__ gi) {
  const int lane = threadIdx.x & 31, w = threadIdx.x >> 5;
  const int lh = lane >> 4, lm = lane & 15;
  const int row0 = blockIdx.y * 128 + w * 16;
  const int col0 = blockIdx.x * 64;
  const int N = 3 * H_;

  v8f acc[4];
#pragma unroll
  for (int j = 0; j < 4; ++j) acc[j] = vzero8();

  const float* arow = hsin + (size_t)(row0 + lm) * H_ + 2 * lh;
  const float* bbase = Wihp + (size_t)lh * 2 * N + 2 * (col0 + lm);

  v2f a0, a1, b0[4], b1[4];
  a0 = *(const v2f*)(arow);
#pragma unroll
  for (int j = 0; j < 4; ++j) b0[j] = *(const v2f*)(bbase + j * 32);

  int k = 0;
  for (; k + 8 < H_; k += 8) {
    const float* bn1 = bbase + (size_t)((k + 4) >> 1) * (2 * N);
    a1 = *(const v2f*)(arow + k + 4);
#pragma unroll
    for (int j = 0; j < 4; ++j) b1[j] = *(const v2f*)(bn1 + j * 32);
#pragma unroll
    for (int j = 0; j < 4; ++j) acc[j] = wmma4(a0, b0[j], acc[j]);
    const float* bn0 = bbase + (size_t)((k + 8) >> 1) * (2 * N);
    a0 = *(const v2f*)(arow + k + 8);
#pragma unroll
    for (int j = 0; j < 4; ++j) b0[j] = *(const v2f*)(bn0 + j * 32);
#pragma unroll
    for (int j = 0; j < 4; ++j) acc[j] = wmma4(a1, b1[j], acc[j]);
  }
  {
    const float* bn1 = bbase + (size_t)((k + 4) >> 1) * (2 * N);
    a1 = *(const v2f*)(arow + k + 4);
#pragma unroll
    for (int j = 0; j < 4; ++j) b1[j] = *(const v2f*)(bn1 + j * 32);
#pragma unroll
    for (int j = 0; j < 4; ++j) acc[j] = wmma4(a0, b0[j], acc[j]);
#pragma unroll
    for (int j = 0; j < 4; ++j) acc[j] = wmma4(a1, b1[j], acc[j]);
  }

#pragma unroll
  for (int j = 0; j < 4; ++j) {
    const int col = col0 + j * 16 + lm;
    const float bv = bih[col];
#pragma unroll
    for (int i = 0; i < 8; ++i) {
      const int row = row0 + i + 8 * lh;
      gi[(size_t)row * N + col] = acc[j][i] + bv;
    }
  }
}

// ---------------------------------------------------------------------------
// K6: persistent GRU recurrence.  1 workgroup, 1024 threads, h in LDS (64KB).
// Per step: gh = h@W_hh (+b_hh); r,z = sigmoid(gi+gh); n = tanh(gi_n + r*gh_n);
// h = (1-z)*n + z*h; store hs2[t].
// ---------------------------------------------------------------------------
__global__ __launch_bounds__(1024) void k_gru_seq(const float* __restrict__ gi,
                                                  const float* __restrict__ Whhp,
                                                  const float* __restrict__ bhh,
                                                  float* __restrict__ hs2) {
  extern __shared__ float smem[];
  float* sh = smem;  // 64 x 256

  const int tid = threadIdx.x;
  const int lane = tid & 31, w = tid >> 5;
  const int lh = lane >> 4, lm = lane & 15;
  const int m_t = w >> 3, oct = w & 7;
  const int row0 = m_t * 16;
  const int N = 3 * H_;

  for (int e = tid; e < B_ * H_; e += 1024) sh[e] = 0.0f;
  __syncthreads();

  const int c0 = (oct * 2) * 16 + lm;
  const float* wb = Whhp + (size_t)lh * 2 * N + 2 * c0;

  for (int t = 0; t < T_; ++t) {
    const float* gi_t = gi + (size_t)t * (B_ * 3 * H_);

    v8f racc[2], zacc[2], nacc[2];
    racc[0] = racc[1] = zacc[0] = zacc[1] = nacc[0] = nacc[1] = vzero8();

    const float* arow = sh + (row0 + lm) * H_ + 2 * lh;
    v2f a0, a1, br0[2], bz0[2], bn0_[2], br1[2], bz1[2], bn1_[2];
    a0 = *(const v2f*)(arow);
#pragma unroll
    for (int j = 0; j < 2; ++j) {
      br0[j]  = *(const v2f*)(wb + j * 32);
      bz0[j]  = *(const v2f*)(wb + j * 32 + 512);    // +H_ cols * 2
      bn0_[j] = *(const v2f*)(wb + j * 32 + 1024);   // +2H_ cols * 2
    }
    int k = 0;
    for (; k + 8 < H_; k += 8) {
      const float* p1 = wb + (size_t)((k + 4) >> 1) * (2 * N);
      a1 = *(const v2f*)(arow + k + 4);
#pragma unroll
      for (int j = 0; j < 2; ++j) {
        br1[j]  = *(const v2f*)(p1 + j * 32);
        bz1[j]  = *(const v2f*)(p1 + j * 32 + 512);
        bn1_[j] = *(const v2f*)(p1 + j * 32 + 1024);
      }
#pragma unroll
      for (int j = 0; j < 2; ++j) {
        racc[j] = wmma4(a0, br0[j], racc[j]);
        zacc[j] = wmma4(a0, bz0[j], zacc[j]);
        nacc[j] = wmma4(a0, bn0_[j], nacc[j]);
      }
      const float* p0 = wb + (size_t)((k + 8) >> 1) * (2 * N);
      a0 = *(const v2f*)(arow + k + 8);
#pragma unroll
      for (int j = 0; j < 2; ++j) {
        br0[j]  = *(const v2f*)(p0 + j * 32);
        bz0[j]  = *(const v2f*)(p0 + j * 32 + 512);
        bn0_[j] = *(const v2f*)(p0 + j * 32 + 1024);
      }
#pragma unroll
      for (int j = 0; j < 2; ++j) {
        racc[j] = wmma4(a1, br1[j], racc[j]);
        zacc[j] = wmma4(a1, bz1[j], zacc[j]);
        nacc[j] = wmma4(a1, bn1_[j], nacc[j]);
      }
    }
    {
      const float* p1 = wb + (size_t)((k + 4) >> 1) * (2 * N);
      a1 = *(const v2f*)(arow + k + 4);
#pragma unroll
      for (int j = 0; j < 2; ++j) {
        br1[j]  = *(const v2f*)(p1 + j * 32);
        bz1[j]  = *(const v2f*)(p1 + j * 32 + 512);
        bn1_[j] = *(const v2f*)(p1 + j * 32 + 1024);
      }
#pragma unroll
      for (int j = 0; j < 2; ++j) {
        racc[j] = wmma4(a0, br0[j], racc[j]);
        zacc[j] = wmma4(a0, bz0[j], zacc[j]);
        nacc[j] = wmma4(a0, bn0_[j], nacc[j]);
      }
#pragma unroll
      for (int j = 0; j < 2; ++j) {
        racc[j] = wmma4(a1, br1[j], racc[j]);
        zacc[j] = wmma4(a1, bz1[j], zacc[j]);
        nacc[j] = wmma4(a1, bn1_[j], nacc[j]);
      }
    }

    float hnew[2][8];
#pragma unroll
    for (int j = 0; j < 2; ++j) {
      const int cc = (oct * 2 + j) * 16 + lm;
      const float bhr = bhh[cc];
      const float bhz = bhh[H_ + cc];
      const float bhn = bhh[2 * H_ + cc];
#pragma unroll
      for (int i = 0; i < 8; ++i) {
        const int m = row0 + i + 8 * lh;
        const float gr = gi_t[(size_t)m * N + cc];
        const float gz = gi_t[(size_t)m * N + H_ + cc];
        const float gn = gi_t[(size_t)m * N + 2 * H_ + cc];
        const float rv = sigmoidf_(gr + racc[j][i] + bhr);
        const float zv = sigmoidf_(gz + zacc[j][i] + bhz);
        const float nv = tanhf(gn + rv * (nacc[j][i] + bhn));
        const float hv = sh[m * H_ + cc];
        hnew[j][i] = (1.0f - zv) * nv + zv * hv;
      }
    }
    __syncthreads();
    float* hs2_t = hs2 + (size_t)t * (B_ * H_);
#pragma unroll
    for (int j = 0; j < 2; ++j) {
      const int cc = (oct * 2 + j) * 16 + lm;
#pragma unroll
      for (int i = 0; i < 8; ++i) {
        const int m = row0 + i + 8 * lh;
        sh[m * H_ + cc] = hnew[j][i];
        hs2_t[(size_t)m * H_ + cc] = hnew[j][i];
      }
    }
    __syncthreads();
  }
}

// ---------------------------------------------------------------------------
// K7: out[b,t] = sigmoid(hs2[t,b,:] . W_out + b_out).  One wave per row.
// ---------------------------------------------------------------------------
__global__ __launch_bounds__(256) void k_out(const float* __restrict__ hs2,
                                             const float* __restrict__ Wout,
                                             const float* __restrict__ bout,
                                             float* __restrict__ out) {
  const int lane = threadIdx.x & 31, w = threadIdx.x >> 5;
  const int r = blockIdx.x * 8 + w;  // 0 .. B*T-1, time-major
  const float* hrow = hs2 + (size_t)r * H_;
  float s = 0.0f;
#pragma unroll
  for (int q = 0; q < 8; ++q) s += hrow[lane + 32 * q] * Wout[lane + 32 * q];
#pragma unroll
  for (int off = 16; off > 0; off >>= 1) s += __shfl_xor(s, off, 32);
  if (lane == 0) {
    const int t = r >> 6, b = r & 63;
    out[(size_t)b * T_ + t] = sigmoidf_(s + bout[0]);
  }
}

// ---------------------------------------------------------------------------
extern "C" void kernel_launch(void* const* d_in, const int* in_sizes, int n_in,
                              void* d_out, int out_size, void* d_ws,
                              size_t ws_size, hipStream_t stream) {
  (void)in_sizes; (void)n_in; (void)out_size; (void)ws_size;
  const float* x      = (const float*)d_in[0];
  const float* mask   = (const float*)d_in[1];
  const float* delta  = (const float*)d_in[2];
  const float* x_mean = (const float*)d_in[3];
  const float* w_dg_x = (const float*)d_in[4];
  const float* b_dg_x = (const float*)d_in[5];
  const float* w_dg_h = (const float*)d_in[6];
  const float* b_dg_h = (const float*)d_in[7];
  const float* W_x    = (const float*)d_in[8];
  const float* U_zr   = (const float*)d_in[9];
  const float* U_n    = (const float*)d_in[10];
  const float* W_m    = (const float*)d_in[11];
  const float* b_g    = (const float*)d_in[12];
  const float* W_ih   = (const float*)d_in[13];
  const float* W_hh   = (const float*)d_in[14];
  const float* b_ih   = (const float*)d_in[15];
  const float* b_hh   = (const float*)d_in[16];
  const float* W_out  = (const float*)d_in[17];
  const float* b_out  = (const float*)d_in[18];
  float* out = (float*)d_out;

  // Workspace layout (floats). ~195.4MB total; gi reuses pre, hs2 reuses the
  // gamma region (both dead by the time they're overwritten).
  float* ws = (float*)d_ws;
  const size_t RT = (size_t)B_ * T_;  // 32768 rows
  float* gx  = ws;                    // RT * F
  float* gh  = gx + RT * F_;          // RT * H
  float* xh  = gh + RT * H_;          // RT * F
  float* pre = xh + RT * F_;          // RT * 3H  (reused as gi)
  float* hs  = pre + RT * 3 * H_;     // RT * H
  float* hs2 = ws;                    // RT * H   (reuses gx + part of gh)
  // packed weights (WMMA B-fragment order)
  float* p   = hs + RT * H_;
  float* wdgx_p = p;            p += (size_t)F_ * F_;
  float* wdgh_p = p;            p += (size_t)F_ * H_;
  float* Wx_p   = p;            p += (size_t)F_ * 3 * H_;
  float* Wm_p   = p;            p += (size_t)F_ * 3 * H_;
  float* Uzr_p  = p;            p += (size_t)H_ * 2 * H_;
  float* Un_p   = p;            p += (size_t)H_ * H_;
  float* Wih_p  = p;            p += (size_t)H_ * 3 * H_;
  float* Whh_p  = p;            p += (size_t)H_ * 3 * H_;

  const dim3 blk(256);
  // one-time weight repacks (tiny)
  k_repack<<<dim3((F_ * F_) / 256),     blk, 0, stream>>>(w_dg_x, wdgx_p, F_, F_);
  k_repack<<<dim3((F_ * H_) / 256),     blk, 0, stream>>>(w_dg_h, wdgh_p, F_, H_);
  k_repack<<<dim3((F_ * 3 * H_) / 256), blk, 0, stream>>>(W_x,  Wx_p,  F_, 3 * H_);
  k_repack<<<dim3((F_ * 3 * H_) / 256), blk, 0, stream>>>(W_m,  Wm_p,  F_, 3 * H_);
  k_repack<<<dim3((H_ * 2 * H_) / 256), blk, 0, stream>>>(U_zr, Uzr_p, H_, 2 * H_);
  k_repack<<<dim3((H_ * H_) / 256),     blk, 0, stream>>>(U_n,  Un_p,  H_, H_);
  k_repack<<<dim3((H_ * 3 * H_) / 256), blk, 0, stream>>>(W_ih, Wih_p, H_, 3 * H_);
  k_repack<<<dim3((H_ * 3 * H_) / 256), blk, 0, stream>>>(W_hh, Whh_p, H_, 3 * H_);

  k_gamma<<<dim3(F_ / 64, (unsigned)(RT / 128)), blk, 0, stream>>>(
      delta, wdgx_p, b_dg_x, gx, F_);
  k_gamma<<<dim3(H_ / 64, (unsigned)(RT / 128)), blk, 0, stream>>>(
      delta, wdgh_p, b_dg_h, gh, H_);
  k_xhat<<<dim3((B_ * F_) / 256), blk, 0, stream>>>(x, mask, gx, x_mean, xh);
  k_pre<<<dim3((3 * H_) / 64, (unsigned)(RT / 128)), blk, 0, stream>>>(
      xh, mask, Wx_p, Wm_p, b_g, pre);
  k_grud_seq<<<1, 1024, 2 * B_ * H_ * sizeof(float), stream>>>(pre, gh, Uzr_p,
                                                               Un_p, hs);
  k_gi<<<dim3((3 * H_) / 64, (unsigned)(RT / 128)), blk, 0, stream>>>(
      hs, Wih_p, b_ih, pre);
  k_gru_seq<<<1, 1024, B_ * H_ * sizeof(float), stream>>>(pre, Whh_p, b_hh, hs2);
  k_out<<<dim3((unsigned)(RT / 8)), blk, 0, stream>>>(hs2, W_out, b_out, out);
}